// ResidualQuantizer_57475252355773
// MI455X (gfx1250) — compile-verified
//
#include <hip/hip_runtime.h>
#include <hip/hip_bf16.h>

typedef __attribute__((ext_vector_type(2))) float v2f;
typedef __attribute__((ext_vector_type(8))) float v8f;

// Typed pointers for the gfx1250 async global->LDS builtin:
// param 0: AS1 (global) int __vector(4)*, param 1: AS3 (LDS) int __vector(4)*.
typedef int rvq_v4i __attribute__((vector_size(16)));
typedef __attribute__((address_space(1))) rvq_v4i* rvq_gv4p;
typedef __attribute__((address_space(3))) rvq_v4i* rvq_lv4p;

// Problem constants (from reference)
#define RVQ_B   32
#define RVQ_C   256
#define RVQ_HW  1024                     // 32*32
#define RVQ_N   32768                    // B*HW
#define RVQ_K   2048
#define RVQ_D   4
#define RVQ_S   8388608L                 // B*C*HW elements per depth plane

// gfx1250 async global->LDS staging (no VGPR round-trip, tracked on ASYNCcnt).
#if defined(__gfx1250__) && \
    __has_builtin(__builtin_amdgcn_global_load_async_to_lds_b128) && \
    __has_builtin(__builtin_amdgcn_s_wait_asynccnt)
#define RVQ_ASYNC_STAGE 1
#else
#define RVQ_ASYNC_STAGE 0
#endif

// ---------------------------------------------------------------------------
// Init: code norms ||e||^2 per codebook row + zero the histogram bins.
// ---------------------------------------------------------------------------
__global__ __launch_bounds__(256) void rvq_init_kernel(
    const float* __restrict__ codebook, float* __restrict__ eNorm,
    int* __restrict__ counts) {
  int k = blockIdx.x * 256 + threadIdx.x;        // 0..2047
  const float* row = codebook + (long)k * RVQ_C;
  float s = 0.f;
  for (int c = 0; c < RVQ_C; c += 4) {
    float4 v = *(const float4*)(row + c);
    s += v.x * v.x + v.y * v.y + v.z * v.z + v.w * v.w;
  }
  eNorm[k] = s;
  counts[k] = 0;
}

// ---------------------------------------------------------------------------
// Argmin over the codebook. Block = 4 wave32; each wave owns one 16-row
// residual tile (64 rows/block). Codebook tiles (16 codes x 256) are staged
// into LDS once per block and shared by all waves, double-buffered: the
// async global->LDS stream for ct+1 overlaps the 64-WMMA chain on ct.
// residual = z - qcum[depth-1] (qcum lives in d_out => no residual scratch).
// D = A(16x4) x B(4x16), K stepped by 4 via V_WMMA_F32_16X16X4_F32.
// ---------------------------------------------------------------------------
__global__ __launch_bounds__(128) void rvq_argmin_kernel(
    const float* __restrict__ z, const float* __restrict__ qcum_prev,
    const float* __restrict__ codebook, const float* __restrict__ eNorm,
    int* __restrict__ idxBuf, float* __restrict__ idxOutF, int depth) {
  constexpr int LDA = 260;   // 260 % 64 == 4: conflict-free rows; 260*4 % 16 == 0
  __shared__ float As[4 * 16 * LDA];             // 66.5 KB: one tile per wave
  __shared__ float Bs[2 * 16 * LDA];             // 33.3 KB: double-buffered codes

  const int t    = threadIdx.x;                  // 0..127
  const int lane = t & 31;
  const int wave = t >> 5;                       // 0..3
  const int n0   = blockIdx.x * 64 + wave * 16;  // this wave's 16 rows

  // Per-wave residual tile load (NHWC gather from NCHW layout).
  // i = col*16 + row: consecutive lanes hit consecutive hw -> coalesced.
  {
    float* Aw = As + wave * (16 * LDA);
    for (int i = lane; i < 16 * RVQ_C; i += 32) {
      int row = i & 15, col = i >> 4;
      int n = n0 + row;                          // 1024 % 16 == 0: stays in one b
      int b = n >> 10, hw = n & 1023;
      long addr = (long)b * (RVQ_C * RVQ_HW) + (long)col * RVQ_HW + hw;
      float r = z[addr];
      if (depth > 0) r -= qcum_prev[addr];
      Aw[row * LDA + col] = r;
    }
  }

  // Cooperative codebook tile staging: 16 codes x 256 f32.
  auto stageB = [&](int buf, int ct) {
    const float4* src = (const float4*)(codebook + (long)ct * 16 * RVQ_C);
    float* dst = Bs + buf * (16 * LDA);
    for (int i = t; i < 16 * RVQ_C / 4; i += 128) {   // 1024 float4
      int code = i >> 6;
      int col4 = (i & 63) << 2;
#if RVQ_ASYNC_STAGE
      // Direct global->LDS copy, no VGPR staging; completion on ASYNCcnt.
      __builtin_amdgcn_global_load_async_to_lds_b128(
          (rvq_gv4p)(src + i),
          (rvq_lv4p)(dst + code * LDA + col4),
          0, 0);
#else
      *(float4*)(dst + code * LDA + col4) = src[i];
#endif
    }
  };

  stageB(0, 0);
#if RVQ_ASYNC_STAGE
  __builtin_amdgcn_s_wait_asynccnt(0);
#endif
  __syncthreads();

  const int jloc  = lane & 15;                   // N-column / M-row within 16
  const int khalf = lane >> 4;                   // lanes 16-31 hold K=2,3

  float bestD[8];
  int   bestI[8];
#pragma unroll
  for (int v = 0; v < 8; ++v) { bestD[v] = 3.4e38f; bestI[v] = 0x7fffffff; }

  const float* arow = As + wave * (16 * LDA) + jloc * LDA + khalf * 2;

  for (int ct = 0; ct < RVQ_K / 16; ++ct) {
    const int cur = ct & 1;
    if (ct + 1 < RVQ_K / 16) stageB(cur ^ 1, ct + 1);  // overlap with WMMA chain

    const float* brow = Bs + cur * (16 * LDA) + jloc * LDA + khalf * 2;
    v8f acc = {};
#pragma unroll 8
    for (int kc = 0; kc < RVQ_C / 4; ++kc) {
      v2f a = *(const v2f*)(arow + kc * 4);      // A: lane=M row, VGPRs K pair
      v2f b = *(const v2f*)(brow + kc * 4);      // B: lane=N col, VGPRs K pair
      acc = __builtin_amdgcn_wmma_f32_16x16x4_f32(
          false, a, false, b, (short)0, acc, false, false);
    }

    float en   = eNorm[ct * 16 + jloc];
    int   myIx = ct * 16 + jloc;
#pragma unroll
    for (int v = 0; v < 8; ++v) {                // lane holds rows v and 8+v
      float dist = en - 2.0f * acc[v];           // ||e||^2 - 2 z.e (||z||^2 const)
      if (dist < bestD[v]) { bestD[v] = dist; bestI[v] = myIx; }  // ct ascending
    }

#if RVQ_ASYNC_STAGE
    __builtin_amdgcn_s_wait_asynccnt(0);         // wave-local staging complete
#endif
    __syncthreads();                             // cross-wave buffer handoff
  }

  // 16-lane xor butterfly per half; every wave saw ALL codes -> final result.
#pragma unroll
  for (int v = 0; v < 8; ++v) {
    float d = bestD[v]; int ix = bestI[v];
#pragma unroll
    for (int off = 8; off >= 1; off >>= 1) {
      float od = __shfl_xor(d, off, 32);
      int   oi = __shfl_xor(ix, off, 32);
      if (od < d || (od == d && oi < ix)) { d = od; ix = oi; }  // argmin: low idx
    }
    if (jloc == 0) {
      int n = n0 + v + 8 * khalf;
      idxBuf [depth * RVQ_N + n] = ix;
      idxOutF[depth * RVQ_N + n] = (float)ix;    // indices output (float dtype)
    }
  }
}

// ---------------------------------------------------------------------------
// Gather chosen code, prefix-sum quantized_cum, emit ST output at last depth.
// ---------------------------------------------------------------------------
__global__ __launch_bounds__(256) void rvq_update_kernel(
    const float* __restrict__ z, const float* __restrict__ codebook,
    const int* __restrict__ idxBuf, float* __restrict__ qcum,
    float* __restrict__ stOut, int depth) {
  long e  = (long)blockIdx.x * 256 + threadIdx.x;  // 0..S-1 over [B,C,H,W]
  int  b  = (int)(e >> 18);                        // / (C*HW)
  int  c  = (int)((e >> 10) & 255);
  int  hw = (int)(e & 1023);
  int  n  = b * RVQ_HW + hw;
  int  ix = idxBuf[depth * RVQ_N + n];
  float q    = codebook[(long)ix * RVQ_C + c];
  float prev = depth ? qcum[(long)(depth - 1) * RVQ_S + e] : 0.0f;
  float cum  = prev + q;
  qcum[(long)depth * RVQ_S + e] = cum;
  if (depth == RVQ_D - 1) stOut[e] = z[e] - cum;   // q_last + (resid - q_last)
}

// ---------------------------------------------------------------------------
// Histogram of all D*N indices, then single-block perplexity reduction.
// ---------------------------------------------------------------------------
__global__ __launch_bounds__(256) void rvq_hist_kernel(
    const int* __restrict__ idxBuf, int* __restrict__ counts) {
  int i = blockIdx.x * 256 + threadIdx.x;          // 0..131071
  atomicAdd(&counts[idxBuf[i]], 1);
}

__global__ __launch_bounds__(256) void rvq_perp_kernel(
    const int* __restrict__ counts, float* __restrict__ out) {
  __shared__ float red[256];
  float s = 0.f;
  const float inv = 1.0f / (float)(RVQ_D * RVQ_N);
  for (int k = threadIdx.x; k < RVQ_K; k += 256) {
    float p = (float)counts[k] * inv;
    s += p * logf(fmaxf(p, 1e-10f));               // p==0 -> 0 * log(1e-10) == 0
  }
  red[threadIdx.x] = s;
  __syncthreads();
  for (int off = 128; off; off >>= 1) {
    if (threadIdx.x < off) red[threadIdx.x] += red[threadIdx.x + off];
    __syncthreads();
  }
  if (threadIdx.x == 0) *out = expf(-red[0]);
}

// ---------------------------------------------------------------------------
extern "C" void kernel_launch(void* const* d_in, const int* in_sizes, int n_in,
                              void* d_out, int out_size, void* d_ws, size_t ws_size,
                              hipStream_t stream) {
  const float* z        = (const float*)d_in[0];   // [32,256,32,32]
  const float* codebook = (const float*)d_in[1];   // [2048,256]

  float* out_qcum = (float*)d_out;                 // [4][S]
  float* out_st   = out_qcum + (long)RVQ_D * RVQ_S;// [S]
  float* out_idx  = out_st + RVQ_S;                // [4*N] (as float)
  float* out_perp = out_idx + (long)RVQ_D * RVQ_N; // [1]

  int*   idxBuf = (int*)d_ws;                      // 4*32768 ints
  int*   counts = idxBuf + RVQ_D * RVQ_N;          // 2048 ints
  float* eNorm  = (float*)(counts + RVQ_K);        // 2048 floats

  rvq_init_kernel<<<RVQ_K / 256, 256, 0, stream>>>(codebook, eNorm, counts);

  for (int d = 0; d < RVQ_D; ++d) {
    const float* qprev = (d == 0) ? z : (out_qcum + (long)(d - 1) * RVQ_S);
    rvq_argmin_kernel<<<RVQ_N / 64, 128, 0, stream>>>(
        z, qprev, codebook, eNorm, idxBuf, out_idx, d);
    rvq_update_kernel<<<(int)(RVQ_S / 256), 256, 0, stream>>>(
        z, codebook, idxBuf, out_qcum, out_st, d);
  }

  rvq_hist_kernel<<<(RVQ_D * RVQ_N) / 256, 256, 0, stream>>>(idxBuf, counts);
  rvq_perp_kernel<<<1, 256, 0, stream>>>(counts, out_perp);
}